// GIN_5153960755958
// MI455X (gfx1250) — compile-verified
//
#include <hip/hip_runtime.h>
#include <math.h>

#define NHID 128

typedef __attribute__((ext_vector_type(2))) float v2f;
typedef __attribute__((ext_vector_type(8))) float v8f;

// ---------------------------------------------------------------------------
// Wave-level double-row-tile GEMM: computes two 16x16 f32 tiles
//   D0 = A[rows 0-15 of zsrc] * W[:, col0:col0+16]
//   D1 = A[rows 16-31 of zsrc] * W[:, col0:col0+16]
// zsrc: 32x128 tile in LDS (row-major).  W: 128x128 row-major in global.
// A fragment layout (V_WMMA_F32_16X16X4_F32): lanes 0-15 -> M=lane, K=k,k+1;
// lanes 16-31 -> M=lane-16, K=k+2,k+3  (contiguous pair -> ds_load_b64).
// B fragment: lanes 0-15 -> N=lane, K=k,k+1; lanes 16-31 -> N=lane-16, K=k+2,k+3.
// ---------------------------------------------------------------------------
__device__ __forceinline__ void wave_tile_pair(const float* zsrc,
                                               const float* __restrict__ W,
                                               int col0, int nlane, int hi,
                                               v8f& acc0, v8f& acc1)
{
    const int koff = 2 * hi;
    const int col  = col0 + nlane;
#pragma unroll
    for (int k = 0; k < NHID; k += 4) {
        v2f a0 = *(const v2f*)(zsrc + nlane * NHID + k + koff);
        v2f a1 = *(const v2f*)(zsrc + (nlane + 16) * NHID + k + koff);
        v2f b;
        b.x = W[(k + koff) * NHID + col];
        b.y = W[(k + koff + 1) * NHID + col];
        acc0 = __builtin_amdgcn_wmma_f32_16x16x4_f32(false, a0, false, b,
                                                     (short)0, acc0, false, false);
        acc1 = __builtin_amdgcn_wmma_f32_16x16x4_f32(false, a1, false, b,
                                                     (short)0, acc1, false, false);
    }
}

// ---------------------------------------------------------------------------
// out[n,128] = act(in[n,128] @ W[128,128] + b), act = relu if do_relu
// 32 rows per block, 8 waves, each wave does a 16-col slab (2 row tiles).
// ---------------------------------------------------------------------------
__global__ __launch_bounds__(256)
void k_gemm_node(const float* __restrict__ in, const float* __restrict__ W,
                 const float* __restrict__ b, float* __restrict__ out,
                 int n, int do_relu)
{
    __shared__ float zA[32 * NHID];

    const int tid       = threadIdx.x;
    const int block_row = blockIdx.x * 32;

    // Stage 32x128 input tile into LDS (float4, zero-pad past n).
#pragma unroll
    for (int it = 0; it < 4; ++it) {
        int i4   = tid + it * 256;      // float4 index, 1024 total
        int elem = i4 * 4;
        int r    = elem >> 7;
        int c    = elem & 127;
        int grow = block_row + r;
        float4 v = make_float4(0.f, 0.f, 0.f, 0.f);
        if (grow < n) v = *(const float4*)(in + (size_t)grow * NHID + c);
        *(float4*)(&zA[elem]) = v;
    }
    __syncthreads();

    const int wave  = tid >> 5;
    const int lane  = tid & 31;
    const int nlane = lane & 15;
    const int hi    = lane >> 4;
    const int col0  = wave * 16;
    const int col   = col0 + nlane;

    v8f acc0 = {}, acc1 = {};
    wave_tile_pair(zA, W, col0, nlane, hi, acc0, acc1);

    const float bias = b[col];
#pragma unroll
    for (int r = 0; r < 8; ++r) {
        float v0 = acc0[r] + bias;
        float v1 = acc1[r] + bias;
        if (do_relu) { v0 = fmaxf(v0, 0.f); v1 = fmaxf(v1, 0.f); }
        int row0 = block_row + r + 8 * hi;
        int row1 = block_row + 16 + r + 8 * hi;
        if (row0 < n) out[(size_t)row0 * NHID + col] = v0;
        if (row1 < n) out[(size_t)row1 * NHID + col] = v1;
    }
}

// ---------------------------------------------------------------------------
// Fused GIN MLP: h <- relu( relu((h+agg)@W1 + b1) @ W2 + b2 )
// Intermediate kept in LDS; in-place safe (each block only touches its rows).
// ---------------------------------------------------------------------------
__global__ __launch_bounds__(256)
void k_mlp(float* __restrict__ h, const float* __restrict__ agg,
           const float* __restrict__ W1, const float* __restrict__ b1,
           const float* __restrict__ W2, const float* __restrict__ b2,
           int n)
{
    __shared__ float zA[32 * NHID];   // h + agg
    __shared__ float zM[32 * NHID];   // relu(z@W1 + b1)

    const int tid       = threadIdx.x;
    const int block_row = blockIdx.x * 32;

#pragma unroll
    for (int it = 0; it < 4; ++it) {
        int i4   = tid + it * 256;
        int elem = i4 * 4;
        int r    = elem >> 7;
        int c    = elem & 127;
        int grow = block_row + r;
        float4 v = make_float4(0.f, 0.f, 0.f, 0.f);
        if (grow < n) {
            float4 a = *(const float4*)(h + (size_t)grow * NHID + c);
            float4 g = *(const float4*)(agg + (size_t)grow * NHID + c);
            v.x = a.x + g.x; v.y = a.y + g.y; v.z = a.z + g.z; v.w = a.w + g.w;
        }
        *(float4*)(&zA[elem]) = v;
    }
    __syncthreads();

    const int wave  = tid >> 5;
    const int lane  = tid & 31;
    const int nlane = lane & 15;
    const int hi    = lane >> 4;
    const int col0  = wave * 16;
    const int col   = col0 + nlane;

    // GEMM 1 -> LDS (relu + bias)
    {
        v8f acc0 = {}, acc1 = {};
        wave_tile_pair(zA, W1, col0, nlane, hi, acc0, acc1);
        const float bias = b1[col];
#pragma unroll
        for (int r = 0; r < 8; ++r) {
            zM[(r + 8 * hi) * NHID + col]      = fmaxf(acc0[r] + bias, 0.f);
            zM[(16 + r + 8 * hi) * NHID + col] = fmaxf(acc1[r] + bias, 0.f);
        }
    }
    __syncthreads();

    // GEMM 2 -> global (relu + bias), in-place into h
    {
        v8f acc0 = {}, acc1 = {};
        wave_tile_pair(zM, W2, col0, nlane, hi, acc0, acc1);
        const float bias = b2[col];
#pragma unroll
        for (int r = 0; r < 8; ++r) {
            float v0 = fmaxf(acc0[r] + bias, 0.f);
            float v1 = fmaxf(acc1[r] + bias, 0.f);
            int row0 = block_row + r + 8 * hi;
            int row1 = block_row + 16 + r + 8 * hi;
            if (row0 < n) h[(size_t)row0 * NHID + col] = v0;
            if (row1 < n) h[(size_t)row1 * NHID + col] = v1;
        }
    }
}

// ---------------------------------------------------------------------------
// Edge scatter: agg[dst[e],:] += h[src[e],:].  One wave per edge,
// each lane handles 4 consecutive features (float4 gather + 4 f32 atomics).
// ---------------------------------------------------------------------------
__global__ __launch_bounds__(256)
void k_scatter(const float* __restrict__ h, const int* __restrict__ src,
               const int* __restrict__ dst, float* __restrict__ agg, int nedges)
{
    int t    = blockIdx.x * 256 + threadIdx.x;
    int e    = t >> 5;
    int lane = t & 31;
    if (e >= nedges) return;
    int s = src[e];
    int d = dst[e];
    float4 v = *(const float4*)(h + (size_t)s * NHID + lane * 4);
    float* p = agg + (size_t)d * NHID + lane * 4;
    atomicAdd(p + 0, v.x);
    atomicAdd(p + 1, v.y);
    atomicAdd(p + 2, v.z);
    atomicAdd(p + 3, v.w);
}

// ---------------------------------------------------------------------------
// Graph pooling: pooled[batch[i],:] += h[i,:].  One wave per node.
// ---------------------------------------------------------------------------
__global__ __launch_bounds__(256)
void k_pool(const float* __restrict__ h, const int* __restrict__ batch,
            float* __restrict__ pooled, int n)
{
    int t    = blockIdx.x * 256 + threadIdx.x;
    int node = t >> 5;
    int lane = t & 31;
    if (node >= n) return;
    int g = batch[node];
    float4 v = *(const float4*)(h + (size_t)node * NHID + lane * 4);
    float* p = pooled + (size_t)g * NHID + lane * 4;
    atomicAdd(p + 0, v.x);
    atomicAdd(p + 1, v.y);
    atomicAdd(p + 2, v.z);
    atomicAdd(p + 3, v.w);
}

// ---------------------------------------------------------------------------
// Head: logits = g @ W_out[128,16] + b_out; out = log_softmax(logits).
// One wave32 per graph; lanes 0-15 own one class each.
// ---------------------------------------------------------------------------
__global__ __launch_bounds__(32)
void k_head(const float* __restrict__ g, const float* __restrict__ Wout,
            const float* __restrict__ bout, float* __restrict__ out, int nclass)
{
    int graph = blockIdx.x;
    int lane  = threadIdx.x;
    float logit = -INFINITY;
    if (lane < nclass) {
        float acc = bout[lane];
        for (int k = 0; k < NHID; ++k)
            acc += g[(size_t)graph * NHID + k] * Wout[k * nclass + lane];
        logit = acc;
    }
    float m = logit;
#pragma unroll
    for (int mask = 8; mask >= 1; mask >>= 1)
        m = fmaxf(m, __shfl_xor(m, mask));
    float ex = (lane < nclass) ? __expf(logit - m) : 0.f;
    float s = ex;
#pragma unroll
    for (int mask = 8; mask >= 1; mask >>= 1)
        s += __shfl_xor(s, mask);
    if (lane < nclass)
        out[(size_t)graph * nclass + lane] = (logit - m) - __logf(s);
}

__global__ __launch_bounds__(256)
void k_zero(float* __restrict__ p, long long n)
{
    long long i      = (long long)blockIdx.x * 256 + threadIdx.x;
    long long stride = (long long)gridDim.x * 256;
    for (; i < n; i += stride) p[i] = 0.f;
}

// ---------------------------------------------------------------------------
extern "C" void kernel_launch(void* const* d_in, const int* in_sizes, int n_in,
                              void* d_out, int out_size, void* d_ws, size_t ws_size,
                              hipStream_t stream)
{
    const float* x      = (const float*)d_in[0];
    const int*   eidx   = (const int*)d_in[1];
    const int*   batch  = (const int*)d_in[2];
    const float* W_pre  = (const float*)d_in[3];
    const float* b_pre  = (const float*)d_in[4];
    const float* W1     = (const float*)d_in[5];
    const float* b1     = (const float*)d_in[6];
    const float* W2     = (const float*)d_in[7];
    const float* b2     = (const float*)d_in[8];
    const float* W_post = (const float*)d_in[9];
    const float* b_post = (const float*)d_in[10];
    const float* W_out  = (const float*)d_in[11];
    const float* b_out  = (const float*)d_in[12];
    float*       out    = (float*)d_out;

    const int nnodes  = in_sizes[0] / NHID;
    const int nedges  = in_sizes[1] / 2;
    const int nlayer  = in_sizes[5] / (NHID * NHID);
    const int nclass  = in_sizes[12];            // b_out length
    const int ngraphs = out_size / nclass;

    const int* src = eidx;
    const int* dst = eidx + nedges;

    float* h      = (float*)d_ws;
    float* agg    = h + (size_t)nnodes * NHID;
    float* pooled = agg + (size_t)nnodes * NHID;
    float* gbuf   = pooled + (size_t)ngraphs * NHID;

    const int gemm_blocks    = (nnodes + 31) / 32;
    const int scatter_blocks = (nedges * 32 + 255) / 256;
    const int pool_blocks    = (nnodes * 32 + 255) / 256;

    // pre: h = x @ W_pre + b_pre
    k_gemm_node<<<gemm_blocks, 256, 0, stream>>>(x, W_pre, b_pre, h, nnodes, 0);

    for (int i = 0; i < nlayer; ++i) {
        k_zero<<<2048, 256, 0, stream>>>(agg, (long long)nnodes * NHID);
        k_scatter<<<scatter_blocks, 256, 0, stream>>>(h, src, dst, agg, nedges);
        k_mlp<<<gemm_blocks, 256, 0, stream>>>(h, agg,
                                               W1 + (size_t)i * NHID * NHID, b1 + i * NHID,
                                               W2 + (size_t)i * NHID * NHID, b2 + i * NHID,
                                               nnodes);
    }

    k_zero<<<64, 256, 0, stream>>>(pooled, (long long)ngraphs * NHID);
    k_pool<<<pool_blocks, 256, 0, stream>>>(h, batch, pooled, nnodes);

    // post: g = relu(pooled @ W_post + b_post)
    k_gemm_node<<<(ngraphs + 31) / 32, 256, 0, stream>>>(pooled, W_post, b_post,
                                                         gbuf, ngraphs, 1);

    // out: log_softmax(g @ W_out + b_out)
    k_head<<<ngraphs, 32, 0, stream>>>(gbuf, W_out, b_out, out, nclass);
}